// MmbeddingsDecoderGrowthModel_71622874628175
// MI455X (gfx1250) — compile-verified
//
#include <hip/hip_runtime.h>
#include <hip/hip_bf16.h>

// Segment-mean + gather + sigmoid decode for MmbeddingsDecoderGrowthModel.
// Q = 100000 segments, N = 8,000,000 rows. Memory-bound: ~224 MB min HBM traffic
// (~10 us at 23.3 TB/s) + 32M L2-side f32 atomics (the real fixed cost).
//
// Cache policy (CDNA5 TH hints):
//   - Z_idx: RT on first read (stays resident in 192 MB L2), NT on last read.
//   - mmbeddings / X_input / output: single-touch streams -> NT.
//   - [Q][4] accumulator/B table (1.6 MB): RT everywhere -> L2-hot for the
//     32M atomic adds and 8M b128 gathers.

#define SEG_Q 100000

typedef float f32x4 __attribute__((ext_vector_type(4)));
typedef int   i32x4 __attribute__((ext_vector_type(4)));

// ---------------------------------------------------------------------------
// Kernel 0: zero the Q x 4 accumulator table (sums[3] + count) in workspace.
// Must run every launch (accumulation must start from zero deterministically).
// ---------------------------------------------------------------------------
__global__ __launch_bounds__(256) void k_zero(float* __restrict__ p, int n) {
  int i = blockIdx.x * 256 + threadIdx.x;
  if (i < n) p[i] = 0.0f;
}

// ---------------------------------------------------------------------------
// Kernel 1: scatter-accumulate. Each thread owns 4 consecutive rows:
//   - 1x b128 RT load of indices (keep Z_idx L2-resident for the decode pass)
//   - 3x b128 NT loads of 12 embedding floats (single-touch stream)
//   - 16x global_atomic_add_f32 (no-return) into the L2-resident [Q][4] table
// ---------------------------------------------------------------------------
__global__ __launch_bounds__(256) void k_accum(const int* __restrict__ zidx,
                                               const float* __restrict__ emb,
                                               float* __restrict__ acc,
                                               int n4, int N) {
  int t = blockIdx.x * 256 + threadIdx.x;
  if (t >= n4) return;
  int base = t * 4;
  if (base + 3 < N) {
    i32x4 z = *((const i32x4*)zidx + t);  // RT: want this resident for pass 3
    const f32x4* ep = (const f32x4*)emb + (size_t)t * 3;
    f32x4 e0 = __builtin_nontemporal_load(ep + 0);
    f32x4 e1 = __builtin_nontemporal_load(ep + 1);
    f32x4 e2 = __builtin_nontemporal_load(ep + 2);
    float m[12] = { e0.x, e0.y, e0.z, e0.w,
                    e1.x, e1.y, e1.z, e1.w,
                    e2.x, e2.y, e2.z, e2.w };
#pragma unroll
    for (int j = 0; j < 4; ++j) {
      float* a = acc + (size_t)z[j] * 4;
      __hip_atomic_fetch_add(a + 0, m[3 * j + 0], __ATOMIC_RELAXED, __HIP_MEMORY_SCOPE_AGENT);
      __hip_atomic_fetch_add(a + 1, m[3 * j + 1], __ATOMIC_RELAXED, __HIP_MEMORY_SCOPE_AGENT);
      __hip_atomic_fetch_add(a + 2, m[3 * j + 2], __ATOMIC_RELAXED, __HIP_MEMORY_SCOPE_AGENT);
      __hip_atomic_fetch_add(a + 3, 1.0f,         __ATOMIC_RELAXED, __HIP_MEMORY_SCOPE_AGENT);
    }
  } else {
    // scalar tail (N % 4 != 0 safety; N = 8M is divisible by 4)
    for (int idx = base; idx < N; ++idx) {
      int q = zidx[idx];
      float* a = acc + (size_t)q * 4;
      __hip_atomic_fetch_add(a + 0, emb[3 * (size_t)idx + 0], __ATOMIC_RELAXED, __HIP_MEMORY_SCOPE_AGENT);
      __hip_atomic_fetch_add(a + 1, emb[3 * (size_t)idx + 1], __ATOMIC_RELAXED, __HIP_MEMORY_SCOPE_AGENT);
      __hip_atomic_fetch_add(a + 2, emb[3 * (size_t)idx + 2], __ATOMIC_RELAXED, __HIP_MEMORY_SCOPE_AGENT);
      __hip_atomic_fetch_add(a + 3, 1.0f,                     __ATOMIC_RELAXED, __HIP_MEMORY_SCOPE_AGENT);
    }
  }
}

// ---------------------------------------------------------------------------
// Kernel 2: finalize means in place: B[q] = count>0 ? sum/max(count,1) : 0.
// Keeps the [Q][4] layout so the decode gather is a single aligned b128.
// ---------------------------------------------------------------------------
__global__ __launch_bounds__(256) void k_finalize(float* __restrict__ acc, int q) {
  int i = blockIdx.x * 256 + threadIdx.x;
  if (i >= q) return;
  f32x4 s = *((f32x4*)acc + i);
  float c = s.w;
  float inv = (c > 0.0f) ? (1.0f / fmaxf(c, 1.0f)) : 0.0f;
  f32x4 b;
  b.x = s.x * inv;
  b.y = s.y * inv;
  b.z = s.z * inv;
  b.w = 0.0f;
  *((f32x4*)acc + i) = b;
}

// ---------------------------------------------------------------------------
// Kernel 3: decode. Per thread: 4 rows.
//   - Z_idx: NT b128 (last use; RT copy from pass 1 should still hit L2)
//   - X_input: NT b128 (single touch)
//   - B rows: RT b128 gathers from the 1.6 MB L2-hot table
//   - output: NT b128 store
// ---------------------------------------------------------------------------
__global__ __launch_bounds__(256) void k_decode(const float* __restrict__ x,
                                                const int* __restrict__ zidx,
                                                const float* __restrict__ acc,
                                                const float* __restrict__ b1p,
                                                const float* __restrict__ b2p,
                                                const float* __restrict__ b3p,
                                                float* __restrict__ out,
                                                int n4, int N) {
  const float b1 = *b1p;  // uniform -> scalar loads
  const float b2 = *b2p;
  const float b3 = *b3p;
  int t = blockIdx.x * 256 + threadIdx.x;
  if (t >= n4) return;
  int base = t * 4;
  if (base + 3 < N) {
    i32x4 z  = __builtin_nontemporal_load((const i32x4*)zidx + t);
    f32x4 xv = __builtin_nontemporal_load((const f32x4*)x + t);
    f32x4 o;
#pragma unroll
    for (int j = 0; j < 4; ++j) {
      f32x4 B = *(const f32x4*)(acc + (size_t)z[j] * 4);  // L2-resident gather
      float num   = b1 + B.x;
      float scale = fmaxf(b3 + B.z, 0.1f);
      float ratio = (xv[j] - (b2 + B.y)) / scale;
      float tc    = fminf(fmaxf(-ratio, -50.0f), 50.0f);
      float den   = 1.0f + __expf(tc);
      o[j] = num / den;
    }
    __builtin_nontemporal_store(o, (f32x4*)out + t);
  } else {
    for (int idx = base; idx < N; ++idx) {
      int q = zidx[idx];
      const float* B = acc + (size_t)q * 4;
      float num   = b1 + B[0];
      float scale = fmaxf(b3 + B[2], 0.1f);
      float ratio = (x[idx] - (b2 + B[1])) / scale;
      float tc    = fminf(fmaxf(-ratio, -50.0f), 50.0f);
      out[idx] = num / (1.0f + __expf(tc));
    }
  }
}

// ---------------------------------------------------------------------------
// Launcher. Inputs (setup_inputs order):
//   0: X_input f32[N], 1: Z_idx i32[N], 2: mmbeddings f32[N*3],
//   3: beta_1 f32[1], 4: beta_2 f32[1], 5: beta_3 f32[1]
// Workspace: SEG_Q*4 floats (1.6 MB) accumulator/B table.
// ---------------------------------------------------------------------------
extern "C" void kernel_launch(void* const* d_in, const int* in_sizes, int n_in,
                              void* d_out, int out_size, void* d_ws, size_t ws_size,
                              hipStream_t stream) {
  const float* x   = (const float*)d_in[0];
  const int*   z   = (const int*)d_in[1];
  const float* emb = (const float*)d_in[2];
  const float* b1  = (const float*)d_in[3];
  const float* b2  = (const float*)d_in[4];
  const float* b3  = (const float*)d_in[5];
  float* out = (float*)d_out;
  float* acc = (float*)d_ws;

  const int N    = in_sizes[0];        // rows
  const int accN = SEG_Q * 4;          // accumulator floats
  const int n4   = (N + 3) / 4;        // rows-of-4 per thread

  k_zero<<<(accN + 255) / 256, 256, 0, stream>>>(acc, accN);
  k_accum<<<(n4 + 255) / 256, 256, 0, stream>>>(z, emb, acc, n4, N);
  k_finalize<<<(SEG_Q + 255) / 256, 256, 0, stream>>>(acc, SEG_Q);
  k_decode<<<(n4 + 255) / 256, 256, 0, stream>>>(x, z, acc, b1, b2, b3, out, n4, N);
}